// GAT_D2RL_Critic_33844342293319
// MI455X (gfx1250) — compile-verified
//
#include <hip/hip_runtime.h>
#include <hip/hip_bf16.h>

#define NN 150000
#define NE 4800000
#define INF_ 64
#define HID 16
#define NG 512

typedef __attribute__((ext_vector_type(2))) float v2f;
typedef __attribute__((ext_vector_type(8))) float v8f;

__device__ __forceinline__ float lrelu(float x) { return x > 0.f ? x : 0.2f * x; }
__device__ __forceinline__ unsigned fenc(float f) {
    unsigned u = __float_as_uint(f);
    return (u & 0x80000000u) ? ~u : (u | 0x80000000u);
}
__device__ __forceinline__ float fdec(unsigned u) {
    return __uint_as_float((u & 0x80000000u) ? (u & 0x7fffffffu) : ~u);
}

__global__ void k_zero(float* p, long n) {
    long i = (long)blockIdx.x * blockDim.x + threadIdx.x;
    long s = (long)gridDim.x * blockDim.x;
    for (; i < n; i += s) p[i] = 0.f;
}

// wea = We^T @ a_e  (2 values)
__global__ void k_wea(const float* __restrict__ We, const float* __restrict__ ae, float* wea) {
    if (threadIdx.x == 0) {
        float w0 = 0.f, w1 = 0.f;
        for (int j = 0; j < HID; j++) { w0 += We[j] * ae[j]; w1 += We[HID + j] * ae[j]; }
        wea[0] = w0; wea[1] = w1;
    }
}

// H = X[nrows,K] @ W[K,16] via V_WMMA_F32_16X16X4_F32; one wave per 16-row tile
__global__ void k_gemm_wmma(const float* __restrict__ X, const float* __restrict__ W,
                            float* __restrict__ H, int nrows, int K) {
    int tile = ((int)blockIdx.x * blockDim.x + threadIdx.x) >> 5;
    if (tile * 16 >= nrows) return;                 // wave-uniform guard (EXEC all-1s inside)
    int lane = threadIdx.x & 31;
    int half = lane >> 4, lm = lane & 15;
    long row = (long)tile * 16 + lm;
    v8f acc = {};
    for (int k = 0; k < K; k += 4) {
        int kb = k + 2 * half;
        v2f a, b;
        a[0] = X[row * K + kb];     a[1] = X[row * K + kb + 1];
        b[0] = W[kb * HID + lm];    b[1] = W[(kb + 1) * HID + lm];
        acc = __builtin_amdgcn_wmma_f32_16x16x4_f32(false, a, false, b, (short)0, acc, false, false);
    }
#pragma unroll
    for (int r = 0; r < 8; r++)
        H[((long)tile * 16 + r + 8 * half) * HID + lm] = acc[r];
}

__global__ void k_alpha(const float* __restrict__ H, const float* __restrict__ av_s,
                        const float* __restrict__ av_d, float* __restrict__ alps,
                        float* __restrict__ alpd, int n) {
    int i = blockIdx.x * blockDim.x + threadIdx.x;
    if (i >= n) return;
    float s = 0.f, d = 0.f;
#pragma unroll
    for (int j = 0; j < HID; j++) { float h = H[(long)i * HID + j]; s += h * av_s[j]; d += h * av_d[j]; }
    alps[i] = s; alpd[i] = d;
}

__global__ void k_edge_stats(const int* __restrict__ ei, const float* __restrict__ ea,
                             float* cnt, float* sattr) {
    int e = blockIdx.x * blockDim.x + threadIdx.x;
    if (e >= NE) return;
    int d = ei[NE + e];
    atomicAdd(&cnt[d], 1.f);
    atomicAdd(&sattr[2 * d + 0], ea[2 * (long)e + 0]);
    atomicAdd(&sattr[2 * d + 1], ea[2 * (long)e + 1]);
}

// self-loop logit; seeds segment max (every node has a self loop)
__global__ void k_self_logit(const float* alps, const float* alpd, const float* cnt,
                             const float* sattr, const float* wea,
                             float* slog, unsigned* menc, int n) {
    int i = blockIdx.x * blockDim.x + threadIdx.x;
    if (i >= n) return;
    float cm = fmaxf(cnt[i], 1.f);
    float ae = (sattr[2 * i] * wea[0] + sattr[2 * i + 1] * wea[1]) / cm;
    float l = lrelu(alps[i] + alpd[i] + ae);
    slog[i] = l;
    menc[i] = fenc(l);
}

__global__ void k_edge_max(const int* __restrict__ ei, const float* __restrict__ ea,
                           const float* alps, const float* alpd, const float* wea,
                           float* ebuf, unsigned* menc) {
    int e = blockIdx.x * blockDim.x + threadIdx.x;
    if (e >= NE) return;
    int s = ei[e], d = ei[NE + e];
    float l = lrelu(alps[s] + alpd[d] + ea[2 * (long)e] * wea[0] + ea[2 * (long)e + 1] * wea[1]);
    ebuf[e] = l;
    atomicMax(&menc[d], fenc(l));
}

__global__ void k_self_exp(const float* slog, const unsigned* menc, float* sex,
                           float* denom, int n) {
    int i = blockIdx.x * blockDim.x + threadIdx.x;
    if (i >= n) return;
    float ex = expf(slog[i] - fdec(menc[i]));
    sex[i] = ex;
    atomicAdd(&denom[i], ex);
}

__global__ void k_edge_expsum(const int* __restrict__ ei, float* ebuf,
                              const unsigned* menc, float* denom) {
    int e = blockIdx.x * blockDim.x + threadIdx.x;
    if (e >= NE) return;
    int d = ei[NE + e];
    float ex = expf(ebuf[e] - fdec(menc[d]));
    ebuf[e] = ex;
    atomicAdd(&denom[d], ex);
}

__global__ void k_edge_aggr(const int* __restrict__ ei, const float* __restrict__ ebuf,
                            const float* __restrict__ denom, const float* __restrict__ H,
                            float* agg) {
    int e = blockIdx.x * blockDim.x + threadIdx.x;
    if (e >= NE) return;
    int s = ei[e], d = ei[NE + e];
    float a = ebuf[e] / (denom[d] + 1e-16f);
    const float* hs = H + (long)s * HID;
    float* od = agg + (long)d * HID;
#pragma unroll
    for (int j = 0; j < HID; j++) atomicAdd(&od[j], a * hs[j]);
}

__global__ void k_node_finish(float* agg, const float* H, const float* sex,
                              const float* denom, const float* bias, int n) {
    int i = blockIdx.x * blockDim.x + threadIdx.x;
    if (i >= n) return;
    float a = sex[i] / (denom[i] + 1e-16f);
#pragma unroll
    for (int j = 0; j < HID; j++) {
        float v = agg[(long)i * HID + j] + a * H[(long)i * HID + j] + bias[j];
        agg[(long)i * HID + j] = fmaxf(v, 0.f);
    }
}

__global__ void k_bn_stats(const float* __restrict__ X, float* bnstat, int n) {
    __shared__ float ssum[256], ssq[256];
    int col = threadIdx.x & 15;
    int nl = threadIdx.x >> 4;
    float s = 0.f, q = 0.f;
    for (int i = blockIdx.x * 16 + nl; i < n; i += gridDim.x * 16) {
        float v = X[(long)i * HID + col];
        s += v; q += v * v;
    }
    ssum[threadIdx.x] = s; ssq[threadIdx.x] = q;
    __syncthreads();
    for (int st = 128; st >= 16; st >>= 1) {
        if ((int)threadIdx.x < st) { ssum[threadIdx.x] += ssum[threadIdx.x + st]; ssq[threadIdx.x] += ssq[threadIdx.x + st]; }
        __syncthreads();
    }
    if (threadIdx.x < 16) { atomicAdd(&bnstat[col], ssum[threadIdx.x]); atomicAdd(&bnstat[16 + col], ssq[threadIdx.x]); }
}

__global__ void k_bn_apply(float* X, const float* bnstat, const float* g, const float* b, int n) {
    long i = (long)blockIdx.x * blockDim.x + threadIdx.x;
    if (i >= (long)n * HID) return;
    int col = (int)(i & 15);
    float mu = bnstat[col] / (float)n;
    float var = bnstat[16 + col] / (float)n - mu * mu;
    X[i] = g[col] * (X[i] - mu) * rsqrtf(var + 1e-5f) + b[col];
}

__global__ void k_pool(const float* __restrict__ agg, const int* __restrict__ batch,
                       float* pooled, float* gcnt, int n) {
    int i = blockIdx.x * blockDim.x + threadIdx.x;
    if (i >= n) return;
    int b = batch[i];
    atomicAdd(&gcnt[b], 1.f);
#pragma unroll
    for (int j = 0; j < HID; j++) atomicAdd(&pooled[(long)b * HID + j], agg[(long)i * HID + j]);
}

__global__ void k_head(const float* __restrict__ pooled, const float* __restrict__ gcnt,
                       const float* g1, const float* bb1, const float* Wl1, const float* bl1,
                       const float* g2, const float* bb2, const float* Wl2, const float* bl2,
                       const float* g3, const float* bb3, const float* Wl3, const float* bl3,
                       const float* Wo, const float* bo, float* out) {
    int g = threadIdx.x;  // 512 threads, one graph each
    __shared__ float ssum[32], ssq[32];
    float c = fmaxf(gcnt[g], 1.f);
    float p[16], z[16];
#pragma unroll
    for (int j = 0; j < 16; j++) p[j] = pooled[g * 16 + j] / c;

    // BN1(p) -> Wl1
    {
        float nrm[16];
        if (g < 32) { ssum[g] = 0.f; ssq[g] = 0.f; }
        __syncthreads();
        for (int j = 0; j < 16; j++) { atomicAdd(&ssum[j], p[j]); atomicAdd(&ssq[j], p[j] * p[j]); }
        __syncthreads();
        for (int j = 0; j < 16; j++) {
            float mu = ssum[j] / 512.f, var = ssq[j] / 512.f - mu * mu;
            nrm[j] = g1[j] * (p[j] - mu) * rsqrtf(var + 1e-5f) + bb1[j];
        }
        __syncthreads();
        for (int j = 0; j < 16; j++) {
            float acc = bl1[j];
            for (int k = 0; k < 16; k++) acc += nrm[k] * Wl1[k * 16 + j];
            z[j] = fmaxf(acc, 0.f);
        }
    }
    // BN2([z,p]) -> Wl2
    {
        float cv[32], nrm[32];
        for (int j = 0; j < 16; j++) { cv[j] = z[j]; cv[16 + j] = p[j]; }
        if (g < 32) { ssum[g] = 0.f; ssq[g] = 0.f; }
        __syncthreads();
        for (int j = 0; j < 32; j++) { atomicAdd(&ssum[j], cv[j]); atomicAdd(&ssq[j], cv[j] * cv[j]); }
        __syncthreads();
        for (int j = 0; j < 32; j++) {
            float mu = ssum[j] / 512.f, var = ssq[j] / 512.f - mu * mu;
            nrm[j] = g2[j] * (cv[j] - mu) * rsqrtf(var + 1e-5f) + bb2[j];
        }
        __syncthreads();
        for (int j = 0; j < 16; j++) {
            float acc = bl2[j];
            for (int k = 0; k < 32; k++) acc += nrm[k] * Wl2[k * 16 + j];
            z[j] = fmaxf(acc, 0.f);
        }
    }
    // BN3([z,p]) -> Wl3
    {
        float cv[32], nrm[32];
        for (int j = 0; j < 16; j++) { cv[j] = z[j]; cv[16 + j] = p[j]; }
        if (g < 32) { ssum[g] = 0.f; ssq[g] = 0.f; }
        __syncthreads();
        for (int j = 0; j < 32; j++) { atomicAdd(&ssum[j], cv[j]); atomicAdd(&ssq[j], cv[j] * cv[j]); }
        __syncthreads();
        for (int j = 0; j < 32; j++) {
            float mu = ssum[j] / 512.f, var = ssq[j] / 512.f - mu * mu;
            nrm[j] = g3[j] * (cv[j] - mu) * rsqrtf(var + 1e-5f) + bb3[j];
        }
        __syncthreads();
        for (int j = 0; j < 16; j++) {
            float acc = bl3[j];
            for (int k = 0; k < 32; k++) acc += nrm[k] * Wl3[k * 16 + j];
            z[j] = fmaxf(acc, 0.f);
        }
    }
    float acc = bo[0];
    for (int j = 0; j < 16; j++) acc += z[j] * Wo[j];
    out[g] = acc;
}

extern "C" void kernel_launch(void* const* d_in, const int* in_sizes, int n_in,
                              void* d_out, int out_size, void* d_ws, size_t ws_size,
                              hipStream_t stream) {
    (void)in_sizes; (void)n_in; (void)out_size; (void)ws_size;
    const float* x    = (const float*)d_in[0];
    const int*   ei   = (const int*)d_in[1];
    const float* ea   = (const float*)d_in[2];
    const int*   bat  = (const int*)d_in[3];
    const float* W1   = (const float*)d_in[4];
    const float* We1  = (const float*)d_in[5];
    const float* as1  = (const float*)d_in[6];
    const float* ad1  = (const float*)d_in[7];
    const float* ae1  = (const float*)d_in[8];
    const float* b1   = (const float*)d_in[9];
    const float* bn1g = (const float*)d_in[10];
    const float* bn1b = (const float*)d_in[11];
    const float* W2   = (const float*)d_in[12];
    const float* We2  = (const float*)d_in[13];
    const float* as2  = (const float*)d_in[14];
    const float* ad2  = (const float*)d_in[15];
    const float* ae2  = (const float*)d_in[16];
    const float* b2   = (const float*)d_in[17];
    const float* g1   = (const float*)d_in[18];
    const float* gb1  = (const float*)d_in[19];
    const float* Wl1  = (const float*)d_in[20];
    const float* bl1  = (const float*)d_in[21];
    const float* g2   = (const float*)d_in[22];
    const float* gb2  = (const float*)d_in[23];
    const float* Wl2  = (const float*)d_in[24];
    const float* bl2  = (const float*)d_in[25];
    const float* g3   = (const float*)d_in[26];
    const float* gb3  = (const float*)d_in[27];
    const float* Wl3  = (const float*)d_in[28];
    const float* bl3  = (const float*)d_in[29];
    const float* Wo   = (const float*)d_in[30];
    const float* bo   = (const float*)d_in[31];

    const long N = NN, E = NE;
    float* P = (float*)d_ws;
    float* hbuf  = P;               P += 16 * N;
    float* agg   = P;               P += 16 * N;
    float* alps  = P;               P += N;
    float* alpd  = P;               P += N;
    float* cnt   = P;               P += N;
    float* sattr = P;               P += 2 * N;
    float* slog  = P;               P += N;
    float* sex   = P;               P += N;
    unsigned* menc = (unsigned*)P;  P += N;
    float* denom = P;               P += N;
    float* ebuf  = P;               P += E;
    float* pooled = P;              P += NG * HID;
    float* gcnt  = P;               P += NG;
    float* bnstat = P;              P += 32;
    float* wea   = P;               P += 2;

    const int BLK = 256;
    const int nbN = (NN + BLK - 1) / BLK;
    const int nbE = (NE + BLK - 1) / BLK;
    const int tiles = NN / 16;                 // 9375
    const int nbG = (tiles + 7) / 8;           // 8 waves/block

    // ---- init (ws is poisoned; zero every accumulator each launch) ----
    k_zero<<<1024, BLK, 0, stream>>>(agg, 16 * N);
    k_zero<<<1024, BLK, 0, stream>>>(cnt, N);
    k_zero<<<1024, BLK, 0, stream>>>(sattr, 2 * N);
    k_zero<<<1024, BLK, 0, stream>>>(denom, N);
    k_zero<<<1, 64, 0, stream>>>(bnstat, 32);
    k_zero<<<32, BLK, 0, stream>>>(pooled, NG * HID);
    k_zero<<<2, BLK, 0, stream>>>(gcnt, NG);

    // ---- GAT layer 1 ----
    k_gemm_wmma<<<nbG, BLK, 0, stream>>>(x, W1, hbuf, NN, INF_);
    k_alpha<<<nbN, BLK, 0, stream>>>(hbuf, as1, ad1, alps, alpd, NN);
    k_edge_stats<<<nbE, BLK, 0, stream>>>(ei, ea, cnt, sattr);
    k_wea<<<1, 32, 0, stream>>>(We1, ae1, wea);
    k_self_logit<<<nbN, BLK, 0, stream>>>(alps, alpd, cnt, sattr, wea, slog, menc, NN);
    k_edge_max<<<nbE, BLK, 0, stream>>>(ei, ea, alps, alpd, wea, ebuf, menc);
    k_self_exp<<<nbN, BLK, 0, stream>>>(slog, menc, sex, denom, NN);
    k_edge_expsum<<<nbE, BLK, 0, stream>>>(ei, ebuf, menc, denom);
    k_edge_aggr<<<nbE, BLK, 0, stream>>>(ei, ebuf, denom, hbuf, agg);
    k_node_finish<<<nbN, BLK, 0, stream>>>(agg, hbuf, sex, denom, b1, NN);
    k_bn_stats<<<512, BLK, 0, stream>>>(agg, bnstat, NN);
    k_bn_apply<<<(int)((16 * N + BLK - 1) / BLK), BLK, 0, stream>>>(agg, bnstat, bn1g, bn1b, NN);

    // ---- GAT layer 2 ----
    k_gemm_wmma<<<nbG, BLK, 0, stream>>>(agg, W2, hbuf, NN, HID);
    k_zero<<<1024, BLK, 0, stream>>>(agg, 16 * N);
    k_zero<<<1024, BLK, 0, stream>>>(denom, N);
    k_alpha<<<nbN, BLK, 0, stream>>>(hbuf, as2, ad2, alps, alpd, NN);
    k_wea<<<1, 32, 0, stream>>>(We2, ae2, wea);
    k_self_logit<<<nbN, BLK, 0, stream>>>(alps, alpd, cnt, sattr, wea, slog, menc, NN);
    k_edge_max<<<nbE, BLK, 0, stream>>>(ei, ea, alps, alpd, wea, ebuf, menc);
    k_self_exp<<<nbN, BLK, 0, stream>>>(slog, menc, sex, denom, NN);
    k_edge_expsum<<<nbE, BLK, 0, stream>>>(ei, ebuf, menc, denom);
    k_edge_aggr<<<nbE, BLK, 0, stream>>>(ei, ebuf, denom, hbuf, agg);
    k_node_finish<<<nbN, BLK, 0, stream>>>(agg, hbuf, sex, denom, b2, NN);

    // ---- mean pool + head ----
    k_pool<<<nbN, BLK, 0, stream>>>(agg, bat, pooled, gcnt, NN);
    k_head<<<1, 512, 0, stream>>>(pooled, gcnt,
                                  g1, gb1, Wl1, bl1,
                                  g2, gb2, Wl2, bl2,
                                  g3, gb3, Wl3, bl3,
                                  Wo, bo, (float*)d_out);
}